// ParticleSwarmOptimizer_50964081934650
// MI455X (gfx1250) — compile-verified
//
#include <hip/hip_runtime.h>
#include <math.h>

typedef __attribute__((ext_vector_type(2))) float v2f;
typedef __attribute__((ext_vector_type(8))) float v8f;

// ---------------------------------------------------------------------------
// K0: zero the colsum accumulator (must re-init every call: graph replays
// do not re-poison the workspace).
// ---------------------------------------------------------------------------
__global__ void pso_zero_kernel(float* __restrict__ p, int n) {
  int i = blockIdx.x * blockDim.x + threadIdx.x;
  if (i < n) p[i] = 0.0f;
}

// ---------------------------------------------------------------------------
// K1: per-row L2 norm. One 256-thread block per row; coalesced float4 loads,
// LDS tree reduce, write sqrt(sum-of-squares) to rownorm[row].
// ---------------------------------------------------------------------------
__global__ void pso_rownorm_kernel(const float* __restrict__ x,
                                   float* __restrict__ rownorm, int H) {
  __shared__ float sdata[256];
  const int row = blockIdx.x;
  const int t = threadIdx.x;
  const float4* xr = (const float4*)(x + (size_t)row * H);
  const int n4 = H >> 2;
  float ss = 0.0f;
  for (int i = t; i < n4; i += 256) {
    float4 v = xr[i];
    ss += v.x * v.x + v.y * v.y + v.z * v.z + v.w * v.w;
  }
  sdata[t] = ss;
  __syncthreads();
  for (int s = 128; s > 0; s >>= 1) {
    if (t < s) sdata[t] += sdata[t + s];
    __syncthreads();
  }
  if (t == 0) rownorm[row] = sqrtf(sdata[0]);
}

// ---------------------------------------------------------------------------
// K2: column sums via WMMA:  colsum = ones[1xB] * X[BxH].
// A = all-ones 16x4 f32 (layout-invariant), B-tile = 4 rows x 16 cols of x.
// Each wave owns one 16-column strip and ROWS_PER_WAVE consecutive rows,
// accumulating in the f32 C/D tile; D[M=0, N=lane] (VGPR0, lanes 0..15) is
// the strip's partial column sum. 16 atomics per wave finish the reduction.
// x is already L2-resident after K1 (64 MB << 192 MB L2).
// ---------------------------------------------------------------------------
#define ROWS_PER_WAVE 512

__global__ void pso_colsum_wmma_kernel(const float* __restrict__ x,
                                       float* __restrict__ colsum,
                                       int B, int H, int totalWaves) {
  const int lane = threadIdx.x & 31;
  const int wave = blockIdx.x * (blockDim.x >> 5) + (threadIdx.x >> 5);
  if (wave >= totalWaves) return;  // uniform per wave; EXEC stays all-ones

  const int strips = H >> 4;            // 16 columns per strip
  const int s  = wave % strips;
  const int rc = wave / strips;
  const int n0 = s << 4;
  const int r0 = rc * ROWS_PER_WAVE;

  const int colLane = lane & 15;
  const int krow = (lane >> 4) << 1;    // lanes 0-15 -> rows k+0,k+1; 16-31 -> k+2,k+3

  const float* base = x + (size_t)r0 * H + n0 + colLane;

  v2f a; a[0] = 1.0f; a[1] = 1.0f;      // ones A-matrix
  v8f c0 = {};
  v8f c1 = {};

  for (int j = 0; j < ROWS_PER_WAVE; j += 8) {
    v2f b0, b1;
    b0[0] = base[(size_t)(j + krow) * H];
    b0[1] = base[(size_t)(j + krow + 1) * H];
    b1[0] = base[(size_t)(j + 4 + krow) * H];
    b1[1] = base[(size_t)(j + 4 + krow + 1) * H];
    c0 = __builtin_amdgcn_wmma_f32_16x16x4_f32(false, a, false, b0,
                                               (short)0, c0, false, false);
    c1 = __builtin_amdgcn_wmma_f32_16x16x4_f32(false, a, false, b1,
                                               (short)0, c1, false, false);
  }

  float partial = c0[0] + c1[0];        // D row M=0 (lanes 0..15) = column sums
  if (lane < 16) atomicAdd(&colsum[n0 + lane], partial);
}

// ---------------------------------------------------------------------------
// K3: single-block finalize.
//   fitness = -mean(rownorm); pbf = max(fitness, pbf_old);
//   pstar = first argmax(pbf);  best = PSO position update for pstar only
//   (only pos[argmax] reaches the output, so P*H work collapses to H).
// ---------------------------------------------------------------------------
__global__ void pso_finalize_kernel(const float* __restrict__ pp,
                                    const float* __restrict__ pv,
                                    const float* __restrict__ pbp,
                                    const float* __restrict__ pbf_in,
                                    const float* __restrict__ gbp_in,
                                    const float* __restrict__ gbf_in,
                                    const float* __restrict__ r1,
                                    const float* __restrict__ r2,
                                    const float* __restrict__ w_in,
                                    const float* __restrict__ cw_in,
                                    const float* __restrict__ sw_in,
                                    const float* __restrict__ colsum,
                                    const float* __restrict__ rownorm,
                                    float* __restrict__ best,
                                    int P, int H, int B) {
  __shared__ float sred[512];
  __shared__ int sidx;
  const int t = threadIdx.x;

  // fitness = -sum(rownorm)/B
  float acc = 0.0f;
  for (int i = t; i < B; i += 512) acc += rownorm[i];
  sred[t] = acc;
  __syncthreads();
  for (int s = 256; s > 0; s >>= 1) {
    if (t < s) sred[t] += sred[t + s];
    __syncthreads();
  }
  const float fitness = -sred[0] / (float)B;
  __syncthreads();

  // personal-best fitness + first-occurrence argmax
  float pbf = (t < P) ? fmaxf(fitness, pbf_in[t]) : -INFINITY;
  // jnp.where(fitness > old, fitness, old) == fmaxf for non-NaN inputs
  sred[t] = pbf;
  __syncthreads();
  for (int s = 256; s > 0; s >>= 1) {
    if (t < s) sred[t] = fmaxf(sred[t], sred[t + s]);
    __syncthreads();
  }
  const float mx = sred[0];
  if (t == 0) sidx = 0x7fffffff;
  __syncthreads();
  if (t < P && pbf == mx) atomicMin(&sidx, t);
  __syncthreads();
  const int pstar = sidx;

  const bool imp_star = fitness > pbf_in[pstar];
  const bool g_imp = fitness > gbf_in[0];
  const float w = w_in[0], cw = cw_in[0], sw = sw_in[0];
  const float r1p = r1[pstar], r2p = r2[pstar];
  const float invB = 1.0f / (float)B;

  const float* ppr  = pp  + (size_t)pstar * H;
  const float* pvr  = pv  + (size_t)pstar * H;
  const float* pbpr = pbp + (size_t)pstar * H;

  for (int h = t; h < H; h += 512) {
    const float xm = colsum[h] * invB;
    const float p  = ppr[h];
    const float pb = imp_star ? xm : pbpr[h];
    const float gb = g_imp ? xm : gbp_in[h];
    const float vel = w * pvr[h] + cw * r1p * (pb - p) + sw * r2p * (gb - p);
    best[h] = p + vel;
  }
}

// ---------------------------------------------------------------------------
// K4: out[b,h] = x[b,h] + best[h], float4 streams (x served from L2).
// ---------------------------------------------------------------------------
__global__ void pso_output_kernel(const float4* __restrict__ x4,
                                  const float4* __restrict__ b4,
                                  float4* __restrict__ o4,
                                  unsigned Hq, long long total4) {
  const long long idx = (long long)blockIdx.x * blockDim.x + threadIdx.x;
  if (idx >= total4) return;
  const unsigned h4 = (unsigned)idx % Hq;
  const float4 xv = x4[idx];
  const float4 bv = b4[h4];
  float4 r;
  r.x = xv.x + bv.x;
  r.y = xv.y + bv.y;
  r.z = xv.z + bv.z;
  r.w = xv.w + bv.w;
  o4[idx] = r;
}

// ---------------------------------------------------------------------------
extern "C" void kernel_launch(void* const* d_in, const int* in_sizes, int n_in,
                              void* d_out, int out_size, void* d_ws, size_t ws_size,
                              hipStream_t stream) {
  const float* x   = (const float*)d_in[0];
  const float* w   = (const float*)d_in[1];
  const float* cw  = (const float*)d_in[2];
  const float* sw  = (const float*)d_in[3];
  const float* pp  = (const float*)d_in[4];
  const float* pv  = (const float*)d_in[5];
  const float* pbp = (const float*)d_in[6];
  const float* pbf = (const float*)d_in[7];
  const float* gbp = (const float*)d_in[8];
  const float* gbf = (const float*)d_in[9];
  const float* r1  = (const float*)d_in[10];
  const float* r2  = (const float*)d_in[11];

  const int P = in_sizes[7];          // 512
  const int H = in_sizes[8];          // 4096
  const int B = in_sizes[0] / H;      // 4096

  float* ws      = (float*)d_ws;
  float* colsum  = ws;                // [H]
  float* rownorm = ws + H;            // [B]
  float* best    = ws + H + B;        // [H] (32KB offset -> float4 aligned)

  // K0: zero the atomic accumulator
  pso_zero_kernel<<<(H + 255) / 256, 256, 0, stream>>>(colsum, H);

  // K1: per-row norms
  pso_rownorm_kernel<<<B, 256, 0, stream>>>(x, rownorm, H);

  // K2: column sums on the matrix pipe (ones^T * X via v_wmma_f32_16x16x4_f32)
  const int strips = H >> 4;
  const int rchunks = B / ROWS_PER_WAVE;
  const int totalWaves = strips * rchunks;
  const int blocks = (totalWaves + 7) / 8;   // 8 waves (256 threads) per block
  pso_colsum_wmma_kernel<<<blocks, 256, 0, stream>>>(x, colsum, B, H, totalWaves);

  // K3: fitness, argmax(pbf), best-particle position update
  pso_finalize_kernel<<<1, 512, 0, stream>>>(pp, pv, pbp, pbf, gbp, gbf,
                                             r1, r2, w, cw, sw,
                                             colsum, rownorm, best, P, H, B);

  // K4: out = x + best (broadcast over rows)
  const long long total4 = (long long)B * H / 4;
  const int oblocks = (int)((total4 + 255) / 256);
  pso_output_kernel<<<oblocks, 256, 0, stream>>>((const float4*)x,
                                                 (const float4*)best,
                                                 (float4*)d_out,
                                                 (unsigned)(H / 4), total4);
}